// Decoder_89429809037843
// MI455X (gfx1250) — compile-verified
//
#include <hip/hip_runtime.h>
#include <hip/hip_bf16.h>
#include <math.h>

#define H 400
#define V 18000
#define B 16
#define L 512
#define G 3
#define STEPS 10
#define S 30
#define SB (S*B)

typedef __attribute__((ext_vector_type(2))) float v2f;
typedef __attribute__((ext_vector_type(8))) float v8f;
typedef unsigned int u32x4 __attribute__((ext_vector_type(4)));
typedef unsigned int u32x8 __attribute__((ext_vector_type(8)));
typedef __attribute__((address_space(3))) float lds_f;

// ---------------- wave/block reductions (wave32) ----------------
__device__ __forceinline__ float wave_max(float v) {
#pragma unroll
  for (int o = 16; o > 0; o >>= 1) v = fmaxf(v, __shfl_down(v, o, 32));
  return v;
}
__device__ __forceinline__ float wave_sum(float v) {
#pragma unroll
  for (int o = 16; o > 0; o >>= 1) v += __shfl_down(v, o, 32);
  return v;
}
__device__ __forceinline__ float block_max(float v, float* red) {
  int lane = threadIdx.x & 31, wid = threadIdx.x >> 5, nw = blockDim.x >> 5;
  v = wave_max(v);
  if (lane == 0) red[wid] = v;
  __syncthreads();
  if (wid == 0) {
    float r = (lane < nw) ? red[lane] : -INFINITY;
    r = wave_max(r);
    if (lane == 0) red[0] = r;
  }
  __syncthreads();
  v = red[0];
  __syncthreads();
  return v;
}
__device__ __forceinline__ float block_sum(float v, float* red) {
  int lane = threadIdx.x & 31, wid = threadIdx.x >> 5, nw = blockDim.x >> 5;
  v = wave_sum(v);
  if (lane == 0) red[wid] = v;
  __syncthreads();
  if (wid == 0) {
    float r = (lane < nw) ? red[lane] : 0.f;
    r = wave_sum(r);
    if (lane == 0) red[0] = r;
  }
  __syncthreads();
  v = red[0];
  __syncthreads();
  return v;
}

// ---------------- init: hidden = encoder_hidden (bcast over S), emb = slot embeddings ----------------
__global__ void init_kernel(const float* __restrict__ enc_hidden,
                            const float* __restrict__ slot_emb,
                            const int* __restrict__ domain_idx,
                            const int* __restrict__ slot_idx,
                            float* __restrict__ hidden, float* __restrict__ emb) {
  int idx = blockIdx.x * blockDim.x + threadIdx.x;
  if (idx >= SB * H) return;
  int sb = idx / H, h = idx - sb * H;
  int b = sb & (B - 1), s = sb >> 4;
  hidden[idx] = enc_hidden[b * H + h];
  emb[idx] = slot_emb[domain_idx[s] * H + h] + slot_emb[slot_idx[s] * H + h];
}

// ---------------- fp32 WMMA GEMM: C[M,N] = A[M,K] @ W[N,K]^T (+bias) ----------------
// grid.x = M/16 tiles, grid.y = Ntiles/5, block = 32 (one wave); EXEC all-ones as WMMA requires.
// The 16xK A-strip is DMA'd into LDS by the Tensor Data Mover; W fragments are
// software-pipelined one k-iteration ahead and prefetched 64 elements ahead.
__global__ void __launch_bounds__(32) wmma_gemm_f32(
    const float* __restrict__ A, const float* __restrict__ W,
    const float* __restrict__ bias, float* __restrict__ C, int N, int K) {
  __shared__ float ash[16 * H];   // 25.6 KB A-strip (K == H for all call sites)
  const int NT = 5;
  int tm = blockIdx.x;
  int tn0 = blockIdx.y * NT;
  int lane = threadIdx.x;
  int half = lane >> 4;           // 0: K={0,1}/C rows 0..7 ; 1: K={2,3}/C rows 8..15
  int l16 = lane & 15;

  // --- Tensor Data Mover: 2D tile load (tile_dim0=K elems, tile_dim1=16 rows) ---
  {
    unsigned lds_off = (unsigned)(size_t)(lds_f*)ash;            // LDS byte offset
    unsigned long long ga =
        (unsigned long long)(size_t)(A + (size_t)tm * 16 * K);   // tile global addr
    u32x4 g0;
    g0[0] = 0x1u;                                  // count=1, is_restore=0, gather=0
    g0[1] = lds_off;                               // D#.lds_addr
    g0[2] = (unsigned)(ga & 0xFFFFFFFFu);          // global_addr[31:0]
    g0[3] = (unsigned)((ga >> 32) & 0x01FFFFFFu)   // global_addr[56:32]
            | (2u << 30);                          // type=2 ("image")
    unsigned td0 = (unsigned)K;                    // tensor_dim0 (elements)
    unsigned td1 = 16u;                            // tensor_dim1 (rows)
    unsigned long long st0 = (unsigned long long)K; // tensor_dim0_stride (elements)
    u32x8 g1;
    g1[0] = (2u << 16);                            // data_size=2 (4 bytes); mask=0
    g1[1] = (td0 & 0xFFFFu) << 16;                 // atomic_addr=0 | tensor_dim0[15:0]
    g1[2] = ((td0 >> 16) & 0xFFFFu) | ((td1 & 0xFFFFu) << 16);   // td0[31:16] | td1[15:0]
    g1[3] = ((td1 >> 16) & 0xFFFFu) | ((unsigned)K << 16);       // td1[31:16] | tile_dim0
    g1[4] = 16u;                                   // tile_dim1=16 | tile_dim2=0
    g1[5] = (unsigned)(st0 & 0xFFFFFFFFu);         // dim0_stride[31:0]
    g1[6] = (unsigned)((st0 >> 32) & 0xFFFFu);     // dim0_stride[47:32] | dim1_stride lo=0
    g1[7] = 0u;                                    // dim1_stride hi (unused for 2D)
    asm volatile("tensor_load_to_lds %0, %1" : : "s"(g0), "s"(g1) : "memory");
    __builtin_amdgcn_s_wait_tensorcnt(0);
  }

  const float* wrow[NT];
#pragma unroll
  for (int j = 0; j < NT; ++j)
    wrow[j] = W + (size_t)((tn0 + j) * 16 + l16) * K;

  v8f acc[NT];
  v8f zero = {0.f, 0.f, 0.f, 0.f, 0.f, 0.f, 0.f, 0.f};
#pragma unroll
  for (int j = 0; j < NT; ++j) acc[j] = zero;

  // prologue: W fragments for k=0
  v2f bcur[NT], bnxt[NT];
  int ka0 = half * 2;
#pragma unroll
  for (int j = 0; j < NT; ++j) bcur[j] = (v2f){wrow[j][ka0], wrow[j][ka0 + 1]};

  for (int k = 0; k < K; k += 4) {
    if (k + 4 < K) {                               // pipeline: issue next W loads first
      int kan = k + 4 + half * 2;
#pragma unroll
      for (int j = 0; j < NT; ++j) bnxt[j] = (v2f){wrow[j][kan], wrow[j][kan + 1]};
    }
    if ((k & 63) == 0 && k + 64 < K) {             // global_prefetch_b8 of W strips
#pragma unroll
      for (int j = 0; j < NT; ++j) __builtin_prefetch(wrow[j] + k + 64, 0, 0);
    }
    int ka = k + half * 2;
    v2f a = (v2f){ash[l16 * K + ka], ash[l16 * K + ka + 1]};  // ds_load from TDM strip
#pragma unroll
    for (int j = 0; j < NT; ++j) {
      acc[j] = __builtin_amdgcn_wmma_f32_16x16x4_f32(
          /*neg_a=*/false, a, /*neg_b=*/false, bcur[j],
          /*c_mod=*/(short)0, acc[j], /*reuse_a=*/false, /*reuse_b=*/false);
    }
#pragma unroll
    for (int j = 0; j < NT; ++j) bcur[j] = bnxt[j];
  }
#pragma unroll
  for (int j = 0; j < NT; ++j) {
    int col = (tn0 + j) * 16 + l16;
    float badd = bias ? bias[col] : 0.f;
#pragma unroll
    for (int r = 0; r < 8; ++r) {
      int row = tm * 16 + r + half * 8;            // C layout: VGPR r -> M=r / M=r+8
      C[(size_t)row * N + col] = acc[j][r] + badd;
    }
  }
}

// ---------------- GRU elementwise combine (torch GRUCell semantics) ----------------
__global__ void gru_combine(const float* __restrict__ gi, const float* __restrict__ gh,
                            float* __restrict__ hidden) {
  int idx = blockIdx.x * blockDim.x + threadIdx.x;
  if (idx >= SB * H) return;
  int sb = idx / H, h = idx - sb * H;
  const float* gir = gi + (size_t)sb * 3 * H;
  const float* ghr = gh + (size_t)sb * 3 * H;
  float r = 1.f / (1.f + expf(-(gir[h] + ghr[h])));
  float z = 1.f / (1.f + expf(-(gir[H + h] + ghr[H + h])));
  float n = tanhf(gir[2 * H + h] + r * ghr[2 * H + h]);
  hidden[idx] = (1.f - z) * n + z * hidden[idx];
}

// ---------------- attention scores: scores[s*B+b, l] = <hidden[s,b,:], enc[b,l,:]> ----------------
__global__ void scores_kernel(const float* __restrict__ hidden, const float* __restrict__ enc,
                              const int* __restrict__ cmask, float* __restrict__ scores) {
  __shared__ float hsh[S * H];   // 48 KB
  int b = blockIdx.x;
  int l = blockIdx.y * blockDim.x + threadIdx.x;
  for (int i = threadIdx.x; i < S * H; i += blockDim.x) {
    int s = i / H, h = i - s * H;
    hsh[i] = hidden[(size_t)(s * B + b) * H + h];
  }
  __syncthreads();
  float acc[S];
#pragma unroll
  for (int s = 0; s < S; ++s) acc[s] = 0.f;
  const float* erow = enc + ((size_t)b * L + l) * H;
  for (int h = 0; h < H; ++h) {
    float e = erow[h];
#pragma unroll
    for (int s = 0; s < S; ++s) acc[s] += e * hsh[s * H + h];
  }
  bool valid = cmask[b * L + l] != 0;
#pragma unroll
  for (int s = 0; s < S; ++s)
    scores[(size_t)(s * B + b) * L + l] = valid ? acc[s] : -INFINITY;
}

// ---------------- per-row: softmax over L -> c_probs, context vec, p_gen, gates ----------------
__global__ void context_kernel(const float* __restrict__ hidden, const float* __restrict__ emb,
                               const float* __restrict__ enc, float* __restrict__ scores,
                               const float* __restrict__ w_ratio_w, const float* __restrict__ w_ratio_b,
                               const float* __restrict__ w_gate_w, const float* __restrict__ w_gate_b,
                               float* __restrict__ pgen, float* __restrict__ gates_out) {
  __shared__ float sc[L];
  __shared__ float cvec[H];
  __shared__ float red[8];
  int sb = blockIdx.x, b = sb & (B - 1), tid = threadIdx.x;
  for (int l = tid; l < L; l += blockDim.x) sc[l] = scores[(size_t)sb * L + l];
  __syncthreads();
  float m = -INFINITY;
  for (int l = tid; l < L; l += blockDim.x) m = fmaxf(m, sc[l]);
  m = block_max(m, red);
  float sum = 0.f;
  for (int l = tid; l < L; l += blockDim.x) { float e = expf(sc[l] - m); sc[l] = e; sum += e; }
  sum = block_sum(sum, red);
  float inv = 1.f / sum;
  for (int l = tid; l < L; l += blockDim.x) { float p = sc[l] * inv; sc[l] = p; scores[(size_t)sb * L + l] = p; }
  __syncthreads();
  for (int h = tid; h < H; h += blockDim.x) {
    float acc = 0.f;
    const float* ecol = enc + (size_t)b * L * H + h;
    for (int l = 0; l < L; ++l) acc += sc[l] * ecol[(size_t)l * H];
    cvec[h] = acc;
  }
  __syncthreads();
  const float* hrow = hidden + (size_t)sb * H;
  const float* erow = emb + (size_t)sb * H;
  float part = 0.f;
  for (int h = tid; h < H; h += blockDim.x)
    part += hrow[h] * w_ratio_w[h] + cvec[h] * w_ratio_w[H + h] + erow[h] * w_ratio_w[2 * H + h];
  part = block_sum(part, red);
  if (tid == 0) pgen[sb] = 1.f / (1.f + expf(-(part + w_ratio_b[0])));
  if (gates_out) {
    int s = sb >> 4;
    for (int g = 0; g < G; ++g) {
      float gp = 0.f;
      for (int h = tid; h < H; h += blockDim.x) gp += cvec[h] * w_gate_w[g * H + h];
      gp = block_sum(gp, red);
      if (tid == 0) gates_out[(b * S + s) * G + g] = gp + w_gate_b[g];
    }
  }
}

// ---------------- final: p_context (LDS scatter), softmax over V, p_final, argmax, next emb ----------------
__global__ void final_kernel(const float* __restrict__ logits, const float* __restrict__ cprobs,
                             const float* __restrict__ pgen, const int* __restrict__ ctx_idx,
                             const float* __restrict__ embedding, float* __restrict__ out,
                             int step, float* __restrict__ emb_next) {
  extern __shared__ float pctx[];
  __shared__ float red[8];
  __shared__ float rv[8];
  __shared__ int ri[8];
  __shared__ int bidx_sh;
  int sb = blockIdx.x, b = sb & (B - 1), s = sb >> 4, tid = threadIdx.x;
  const float* lrow = logits + (size_t)sb * V;
  for (int v = tid; v < V; v += blockDim.x) pctx[v] = 0.f;
  __syncthreads();
  for (int l = tid; l < L; l += blockDim.x)
    atomicAdd(&pctx[ctx_idx[b * L + l]], cprobs[(size_t)sb * L + l]);   // ds_add_f32
  __syncthreads();
  float m = -INFINITY;
  for (int v = tid; v < V; v += blockDim.x) m = fmaxf(m, lrow[v]);
  m = block_max(m, red);
  float sum = 0.f;
  for (int v = tid; v < V; v += blockDim.x) sum += expf(lrow[v] - m);
  sum = block_sum(sum, red);
  float inv = 1.f / sum;
  float pg = pgen[sb], qg = 1.f - pg;
  size_t obase = ((size_t)(b * S + s) * STEPS + (size_t)step) * (size_t)V;
  float bestv = -1.f;
  int besti = 0x7fffffff;
  for (int v = tid; v < V; v += blockDim.x) {
    float pv = expf(lrow[v] - m) * inv;
    float pf = qg * pctx[v] + pg * pv;
    out[obase + v] = pf;
    if (pf > bestv) { bestv = pf; besti = v; }   // first-max per thread (ascending v)
  }
#pragma unroll
  for (int o = 16; o > 0; o >>= 1) {
    float ov = __shfl_down(bestv, o, 32);
    int oi = __shfl_down(besti, o, 32);
    if (ov > bestv || (ov == bestv && oi < besti)) { bestv = ov; besti = oi; }
  }
  int lane = tid & 31, wid = tid >> 5;
  if (lane == 0) { rv[wid] = bestv; ri[wid] = besti; }
  __syncthreads();
  if (tid == 0) {
    float bv = rv[0]; int bi = ri[0];
    for (int w = 1; w < (int)(blockDim.x >> 5); ++w)
      if (rv[w] > bv || (rv[w] == bv && ri[w] < bi)) { bv = rv[w]; bi = ri[w]; }
    bidx_sh = bi;
  }
  __syncthreads();
  int pred = bidx_sh;
  for (int h = tid; h < H; h += blockDim.x)
    emb_next[(size_t)sb * H + h] = embedding[(size_t)pred * H + h];   // greedy feedback
}

extern "C" void kernel_launch(void* const* d_in, const int* in_sizes, int n_in,
                              void* d_out, int out_size, void* d_ws, size_t ws_size,
                              hipStream_t stream) {
  const float* enc_hidden = (const float*)d_in[0];
  const float* enc_out    = (const float*)d_in[1];
  const int*   ctx_idx    = (const int*)d_in[2];
  const int*   ctx_mask   = (const int*)d_in[3];
  /* d_in[4] target_values unused (eval / greedy decode) */
  const float* embedding  = (const float*)d_in[5];
  const float* slot_emb   = (const float*)d_in[6];
  const int*   domain_idx = (const int*)d_in[7];
  const int*   slot_idx   = (const int*)d_in[8];
  const float* W_ih       = (const float*)d_in[9];
  const float* W_hh       = (const float*)d_in[10];
  const float* b_ih       = (const float*)d_in[11];
  const float* b_hh       = (const float*)d_in[12];
  const float* w_ratio_w  = (const float*)d_in[13];
  const float* w_ratio_b  = (const float*)d_in[14];
  const float* w_gate_w   = (const float*)d_in[15];
  const float* w_gate_b   = (const float*)d_in[16];

  float* out = (float*)d_out;
  float* gates_out = out + (size_t)B * S * STEPS * V;

  float* ws     = (float*)d_ws;
  float* hidden = ws;  ws += SB * H;
  float* emb    = ws;  ws += SB * H;
  float* gi     = ws;  ws += SB * 3 * H;
  float* gh     = ws;  ws += SB * 3 * H;
  float* logits = ws;  ws += (size_t)SB * V;
  float* cprobs = ws;  ws += SB * L;    // masked scores, then c_probs in place
  float* pgen   = ws;  ws += SB;

  init_kernel<<<(SB * H + 255) / 256, 256, 0, stream>>>(
      enc_hidden, slot_emb, domain_idx, slot_idx, hidden, emb);

  for (int t = 0; t < STEPS; ++t) {
    // GRU gate GEMMs: [480,400] x [400,1200] each, fp32 WMMA + TDM-staged A
    wmma_gemm_f32<<<dim3(SB / 16, (3 * H) / 80), 32, 0, stream>>>(emb,    W_ih, b_ih, gi, 3 * H, H);
    wmma_gemm_f32<<<dim3(SB / 16, (3 * H) / 80), 32, 0, stream>>>(hidden, W_hh, b_hh, gh, 3 * H, H);
    gru_combine<<<(SB * H + 255) / 256, 256, 0, stream>>>(gi, gh, hidden);
    // vocab projection: [480,400] x [400,18000], fp32 WMMA + TDM-staged A
    wmma_gemm_f32<<<dim3(SB / 16, V / 80), 32, 0, stream>>>(hidden, embedding, nullptr, logits, V, H);
    // attention
    scores_kernel<<<dim3(B, L / 128), 128, 0, stream>>>(hidden, enc_out, ctx_mask, cprobs);
    context_kernel<<<SB, 256, 0, stream>>>(hidden, emb, enc_out, cprobs,
                                           w_ratio_w, w_ratio_b, w_gate_w, w_gate_b,
                                           pgen, (t == 0) ? gates_out : (float*)nullptr);
    // pointer-generator mix, softmax over V, output write, argmax feedback
    final_kernel<<<SB, 256, (size_t)V * sizeof(float), stream>>>(
        logits, cprobs, pgen, ctx_idx, embedding, out, t, emb);
  }
}